// AlignTTSLoss_45165876085018
// MI455X (gfx1250) — compile-verified
//
#include <hip/hip_runtime.h>
#include <hip/hip_bf16.h>
#include <math.h>

#define B_SZ   32
#define TXT    256
#define MELT   2048
#define NMEL   80
#define K2     160
#define NEGF   (-1e30f)

typedef float v2f __attribute__((ext_vector_type(2)));
typedef float v8f __attribute__((ext_vector_type(8)));

// ---------------------------------------------------------------------------
// Kernel 1: build A rows (inv_var | -2*mu*inv_var) and bias = t3 + t4 per (b,j).
// One wave32 per row. Also zeroes the loss accumulator d_out[0].
// ---------------------------------------------------------------------------
__global__ void mdn_prep_kernel(const float* __restrict__ mu_logvar,
                                float* __restrict__ Amat,
                                float* __restrict__ bias,
                                float* __restrict__ loss_out) {
    if (blockIdx.x == 0 && threadIdx.x == 0) loss_out[0] = 0.0f;
    int gtid = blockIdx.x * blockDim.x + threadIdx.x;
    int row  = gtid >> 5;             // one wave per (b,j) row
    int lane = threadIdx.x & 31;
    if (row >= B_SZ * TXT) return;

    const float* src = mu_logvar + (size_t)row * K2;
    float* Ar = Amat + (size_t)row * K2;

    float t34 = 0.0f;
    for (int c = lane; c < NMEL; c += 32) {
        float mu = src[c];
        float lv = src[NMEL + c];
        float iv = expf(-lv);
        Ar[c]        = iv;
        Ar[NMEL + c] = -2.0f * mu * iv;
        t34 += mu * mu * iv + lv;
    }
    // wave32 reduction
    for (int off = 16; off > 0; off >>= 1)
        t34 += __shfl_down(t34, off, 32);
    if (lane == 0) bias[row] = t34;
}

// ---------------------------------------------------------------------------
// Kernel 2: log_prob GEMM via V_WMMA_F32_16X16X4_F32.
// Per wave: one 16(j) x 16(t) output tile, K = 160 in steps of 4.
// Block = 256 threads = 8 waves tiling 16(j) x 128(t). Grid (16,16,32).
// K-loop statically split: k in [0,80) uses melspec^2 rows, k in [80,160)
// uses melspec rows directly -> zero branches in the hot loop, 40 WMMAs.
// ---------------------------------------------------------------------------
__global__ void mdn_logprob_wmma_kernel(const float* __restrict__ Amat,
                                        const float* __restrict__ bias,
                                        const float* __restrict__ melspec,
                                        float* __restrict__ logprob) {
    int b    = blockIdx.z;
    int j0   = blockIdx.y * 16;
    int wave = threadIdx.x >> 5;
    int lane = threadIdx.x & 31;
    int t0   = blockIdx.x * 128 + wave * 16;

    int hi   = lane >> 4;      // 0: lanes 0-15 carry K{0,1}; 1: lanes 16-31 carry K{2,3}
    int l16  = lane & 15;
    int arow = j0 + l16;       // A row this lane feeds
    int tcol = t0 + l16;       // output column this lane owns

    // Per-lane base pointers; unrolled loop uses immediate offsets from these.
    const float* Abase = Amat    + ((size_t)b * TXT + arow) * K2 + 2 * hi;
    const float* Mbase = melspec + (size_t)b * NMEL * MELT
                                 + (size_t)(2 * hi) * MELT + tcol;

    v8f acc = {};

    // ---- first half: B rows are melspec^2 (t1 term) ----
    #pragma unroll
    for (int k = 0; k < NMEL; k += 4) {
        v2f afrag = *(const v2f*)(Abase + k);
        float m0 = Mbase[(size_t)k * MELT];
        float m1 = Mbase[(size_t)(k + 1) * MELT];
        v2f bfrag;
        bfrag[0] = m0 * m0;
        bfrag[1] = m1 * m1;
        acc = __builtin_amdgcn_wmma_f32_16x16x4_f32(
            false, afrag, false, bfrag, (short)0, acc, false, false);
    }

    // ---- second half: B rows are melspec (t2 term, A already carries -2*mu*iv) ----
    #pragma unroll
    for (int k = 0; k < NMEL; k += 4) {
        v2f afrag = *(const v2f*)(Abase + NMEL + k);
        v2f bfrag;
        bfrag[0] = Mbase[(size_t)k * MELT];
        bfrag[1] = Mbase[(size_t)(k + 1) * MELT];
        acc = __builtin_amdgcn_wmma_f32_16x16x4_f32(
            false, afrag, false, bfrag, (short)0, acc, false, false);
    }

    // D layout: lane<16 -> rows M=0..7, lane>=16 -> rows M=8..15, column tcol
    const float scale = -0.5f / (float)NMEL;
    float* Ob = logprob + (size_t)b * TXT * MELT;
    const float* biasb = bias + (size_t)b * TXT;
    #pragma unroll
    for (int m = 0; m < 8; ++m) {
        int row = j0 + hi * 8 + m;
        Ob[(size_t)row * MELT + tcol] = scale * (acc[m] + biasb[row]);
    }
}

// ---------------------------------------------------------------------------
// Kernel 3: forward DP (logaddexp scan) over mel_T. One block per batch,
// thread j owns text position j; neighbor via LDS; lp staged in 256x32 tiles.
// ---------------------------------------------------------------------------
__global__ void mdn_dp_kernel(const float* __restrict__ logprob,
                              const int* __restrict__ text_lengths,
                              const int* __restrict__ mel_lengths,
                              float* __restrict__ loss_out) {
    __shared__ float tile[TXT * 33];   // padded stride 33 to dodge bank conflicts
    __shared__ float colbuf[TXT];

    int b = blockIdx.x;
    int j = threadIdx.x;
    const float* lpb = logprob + (size_t)b * TXT * MELT;
    int tl = text_lengths[b];
    int ml = mel_lengths[b];

    float col  = NEGF;
    float term = 0.0f;
    bool  has  = false;

    for (int t0 = 0; t0 < MELT; t0 += 32) {
        // cooperative coalesced tile load: lp[b, j, t0..t0+31] for all j
        #pragma unroll
        for (int r = 0; r < 32; ++r) {
            int idx = r * TXT + threadIdx.x;
            int jj  = idx >> 5;
            int tt  = idx & 31;
            tile[jj * 33 + tt] = lpb[(size_t)jj * MELT + t0 + tt];
        }
        __syncthreads();

        for (int tt = 0; tt < 32; ++tt) {
            int t = t0 + tt;
            float lpv = tile[j * 33 + tt];
            if (t == 0) {
                col = (j == 0) ? lpv : NEGF;
            } else {
                float shifted = (j == 0) ? NEGF : colbuf[j - 1];
                float mx = fmaxf(col, shifted);
                float d  = fminf(col, shifted) - mx;
                col = mx + log1pf(__expf(d)) + lpv;
            }
            if (t == ml - 1 && j == tl - 1) {
                term = col / (float)ml;
                has = true;
            }
            __syncthreads();          // WAR: all reads of colbuf done
            colbuf[j] = col;
            __syncthreads();          // RAW: colbuf visible for next step
        }
    }

    if (has) atomicAdd(loss_out, -term / (float)B_SZ);
}

// ---------------------------------------------------------------------------
extern "C" void kernel_launch(void* const* d_in, const int* in_sizes, int n_in,
                              void* d_out, int out_size, void* d_ws, size_t ws_size,
                              hipStream_t stream) {
    const float* mu_logvar    = (const float*)d_in[0];
    const float* melspec      = (const float*)d_in[1];
    const int*   text_lengths = (const int*)d_in[2];
    const int*   mel_lengths  = (const int*)d_in[3];

    float* out     = (float*)d_out;          // [0] = loss, [1..] = log_prob (B,txt_T,mel_T)
    float* logprob = out + 1;

    float* Amat = (float*)d_ws;                          // B*TXT*K2 floats (~5.25 MB)
    float* bias = Amat + (size_t)B_SZ * TXT * K2;        // B*TXT floats

    // 1) prep: A rows + bias, zero loss accumulator
    {
        int waves   = B_SZ * TXT;            // one wave per row
        int threads = waves * 32;
        mdn_prep_kernel<<<(threads + 255) / 256, 256, 0, stream>>>(
            mu_logvar, Amat, bias, out);
    }

    // 2) log_prob GEMM (WMMA f32 16x16x4)
    {
        dim3 grid(MELT / 128, TXT / 16, B_SZ);
        mdn_logprob_wmma_kernel<<<grid, 256, 0, stream>>>(Amat, bias, melspec, logprob);
    }

    // 3) forward logaddexp scan + loss
    mdn_dp_kernel<<<B_SZ, TXT, 0, stream>>>(logprob, text_lengths, mel_lengths, out);
}